// GNNNet_80582176407957
// MI455X (gfx1250) — compile-verified
//
#include <hip/hip_runtime.h>
#include <hip/hip_bf16.h>

typedef float v2f __attribute__((ext_vector_type(2)));
typedef float v8f __attribute__((ext_vector_type(8)));

// ---------------------------------------------------------------------------
// Tiled fp32 WMMA GEMM:  Y[n,fo] = X[n,fi] * W[fi,fo]  (+bias, +relu)
// One wave computes one 16x16 tile of Y via V_WMMA_F32_16X16X4_F32.
// Out-of-range A rows / B cols are clamped (their products land only in
// guarded-out D rows/cols); only the K tail needs explicit zero padding.
// mode: 0 = raw, 1 = bias+relu, 2 = bias only
// ---------------------------------------------------------------------------
#define WAVES_PER_BLOCK 8

__global__ __launch_bounds__(256) void linear_wmma_f32(
    const float* __restrict__ X, const float* __restrict__ W,
    const float* __restrict__ bias, float* __restrict__ Y,
    int n, int fi, int fo, int ldy, int mode)
{
    const int lane    = threadIdx.x & 31;
    const int wave    = threadIdx.x >> 5;
    const int rowTile = blockIdx.x * WAVES_PER_BLOCK + wave;
    const int colTile = blockIdx.y;
    const int row0 = rowTile * 16;
    const int col0 = colTile * 16;
    if (row0 >= n) return;                 // wave-uniform exit: EXEC stays full

    const int half = lane >> 4;            // 0: K={0,1}, 1: K={2,3}
    const int m    = lane & 15;
    const int arow = min(row0 + m, n - 1);   // clamped: only feeds guarded D rows
    const int bcol = min(col0 + m, fo - 1);  // clamped: only feeds guarded D cols

    const float* ap = X + (size_t)arow * fi + half * 2;          // A: row, K=kb,kb+1
    const float* bp = W + (size_t)(half * 2) * fo + bcol;        // B: K=kb,kb+1, col
    const size_t bstep = (size_t)4 * fo;

    v8f acc = {};
    const int fi_main = fi & ~3;
    for (int k0 = 0; k0 < fi_main; k0 += 4) {
        v2f a; a.x = ap[0]; a.y = ap[1];
        v2f b; b.x = bp[0]; b.y = bp[fo];
        acc = __builtin_amdgcn_wmma_f32_16x16x4_f32(
            false, a, false, b, (short)0, acc, false, false);
        ap += 4;
        bp += bstep;
    }
    if (fi_main < fi) {                    // K tail (fi % 4 == 2 for this net)
        const int kb = fi_main + half * 2;
        v2f a; a.x = 0.0f; a.y = 0.0f;
        v2f b; b.x = 0.0f; b.y = 0.0f;
        if (kb < fi)     { a.x = ap[0]; b.x = bp[0];  }
        if (kb + 1 < fi) { a.y = ap[1]; b.y = bp[fo]; }
        acc = __builtin_amdgcn_wmma_f32_16x16x4_f32(
            false, a, false, b, (short)0, acc, false, false);
    }

    // C/D layout: VGPR r -> (M=r, N=lane) lanes 0-15, (M=r+8, N=lane-16) lanes 16-31
#pragma unroll
    for (int r = 0; r < 8; ++r) {
        const int orow = row0 + r + half * 8;
        const int ocol = col0 + m;
        if (orow < n && ocol < fo) {
            float v = acc[r];
            if (mode >= 1) v += bias[ocol];
            if (mode == 1) v = fmaxf(v, 0.0f);
            Y[(size_t)orow * ldy + ocol] = v;
        }
    }
}

// ---------------------------------------------------------------------------
// Small utility kernels
// ---------------------------------------------------------------------------
__global__ void zero_f32(float* __restrict__ p, size_t nelem) {
    size_t i = (size_t)blockIdx.x * blockDim.x + threadIdx.x;
    if (i < nelem) p[i] = 0.0f;
}
__global__ void fill1_f32(float* __restrict__ p, int nelem) {
    int i = blockIdx.x * blockDim.x + threadIdx.x;
    if (i < nelem) p[i] = 1.0f;
}
__global__ void deg_accum(const int* __restrict__ dst, int ne, float* __restrict__ deg) {
    int i = blockIdx.x * blockDim.x + threadIdx.x;
    if (i < ne) atomicAdd(&deg[dst[i]], 1.0f);
}
__global__ void rsqrt_inplace(float* __restrict__ p, int nelem) {
    int i = blockIdx.x * blockDim.x + threadIdx.x;
    if (i < nelem) p[i] = rsqrtf(p[i]);   // deg >= 1 always (self loop)
}

// ---------------------------------------------------------------------------
// GCN message passing: acc[dst,f] += H[src,f] * dinv[src]*dinv[dst]
// One block (64 threads) per (edge, 64-feature chunk); coalesced loads/atomics.
// ---------------------------------------------------------------------------
__global__ __launch_bounds__(64) void gcn_scatter(
    const int* __restrict__ src, const int* __restrict__ dst,
    const float* __restrict__ dinv, const float* __restrict__ H,
    float* __restrict__ acc, int ne, int fo)
{
    const int e = blockIdx.x;
    const int f = blockIdx.y * 64 + threadIdx.x;
    if (e >= ne || f >= fo) return;
    const int s = src[e], d = dst[e];
    const float c = dinv[s] * dinv[d];
    atomicAdd(&acc[(size_t)d * fo + f], H[(size_t)s * fo + f] * c);
}

// Y[i,f] = relu(acc[i,f] + H[i,f]*dinv[i]^2 + bias[f])   (self-loop + bias + relu)
__global__ __launch_bounds__(64) void gcn_finalize(
    const float* __restrict__ acc, const float* __restrict__ H,
    const float* __restrict__ dinv, const float* __restrict__ bias,
    float* __restrict__ Y, int n, int fo)
{
    const int i = blockIdx.x;
    const int f = blockIdx.y * 64 + threadIdx.x;
    if (i >= n || f >= fo) return;
    const float di = dinv[i];
    const float v = acc[(size_t)i * fo + f] + H[(size_t)i * fo + f] * di * di + bias[f];
    Y[(size_t)i * fo + f] = fmaxf(v, 0.0f);
}

// ---------------------------------------------------------------------------
// Global mean pool
// ---------------------------------------------------------------------------
__global__ __launch_bounds__(64) void pool_accum(
    const float* __restrict__ X, const int* __restrict__ batch,
    float* __restrict__ sum, float* __restrict__ cnt, int n, int fw)
{
    const int i = blockIdx.x;
    const int f = blockIdx.y * 64 + threadIdx.x;
    if (i >= n) return;
    const int g = batch[i];
    if (f < fw) atomicAdd(&sum[(size_t)g * fw + f], X[(size_t)i * fw + f]);
    if (blockIdx.y == 0 && threadIdx.x == 0) atomicAdd(&cnt[g], 1.0f);
}
__global__ __launch_bounds__(64) void pool_div(
    float* __restrict__ sum, const float* __restrict__ cnt, int fw)
{
    const int g = blockIdx.x;
    const int f = blockIdx.y * 64 + threadIdx.x;
    if (f < fw) sum[(size_t)g * fw + f] /= fmaxf(cnt[g], 1.0f);
}

// ---------------------------------------------------------------------------
// Host-side helpers
// ---------------------------------------------------------------------------
static inline void run_linear(const float* X, const float* W, const float* b, float* Y,
                              int n, int fi, int fo, int ldy, int mode, hipStream_t s)
{
    dim3 g((n + 16 * WAVES_PER_BLOCK - 1) / (16 * WAVES_PER_BLOCK), (fo + 15) / 16);
    linear_wmma_f32<<<g, 256, 0, s>>>(X, W, b, Y, n, fi, fo, ldy, mode);
}

static inline void run_gcn(const float* X, const float* W, const float* b,
                           const int* esrc, const int* edst, int ne,
                           const float* dinv, float* H, float* acc, float* Y,
                           int n, int fi, int fo, hipStream_t s)
{
    run_linear(X, W, nullptr, H, n, fi, fo, fo, /*raw*/0, s);
    size_t tot = (size_t)n * fo;
    zero_f32<<<(unsigned)((tot + 255) / 256), 256, 0, s>>>(acc, tot);
    dim3 gs(ne, (fo + 63) / 64);
    gcn_scatter<<<gs, 64, 0, s>>>(esrc, edst, dinv, H, acc, ne, fo);
    dim3 gf(n, (fo + 63) / 64);
    gcn_finalize<<<gf, 64, 0, s>>>(acc, H, dinv, b, Y, n, fo);
}

extern "C" void kernel_launch(void* const* d_in, const int* in_sizes, int n_in,
                              void* d_out, int out_size, void* d_ws, size_t ws_size,
                              hipStream_t stream)
{
    const float* mol_x     = (const float*)d_in[0];
    const float* pro_x     = (const float*)d_in[1];
    const int*   mol_ei    = (const int*)d_in[2];
    const int*   pro_ei    = (const int*)d_in[3];
    const int*   pro_oei   = (const int*)d_in[4];
    const int*   mol_batch = (const int*)d_in[5];
    const int*   pro_batch = (const int*)d_in[6];
    auto prm = [&](int i) { return (const float*)d_in[7 + i]; };
    // params: 0 mc1_w 1 mc1_b 2 mc2_w 3 mc2_b 4 mc3_w 5 mc3_b 6 mg1_w 7 mg1_b
    //         8 mg2_w 9 mg2_b 10 pc1_w 11 pc1_b 12 pc2_w 13 pc2_b 14 pc3_w 15 pc3_b
    //         16 pg1_w 17 pg1_b 18 pg2_w 19 pg2_b 20 fc1_w 21 fc1_b 22 fc2_w 23 fc2_b
    //         24 out_w 25 out_b

    const int n_mol  = in_sizes[0] / 78;
    const int n_pro  = in_sizes[1] / 54;
    const int ne_mol = in_sizes[2] / 2;
    const int ne_pro = in_sizes[3] / 2;
    const int B = 256;

    // ---- workspace layout (floats) ----
    float* w = (float*)d_ws;
    size_t off = 0;
    auto alloc = [&](size_t nelem) { float* p = w + off; off += nelem; return p; };
    float* pro_dinv1 = alloc(n_pro);
    float* pro_dinv2 = alloc(n_pro);
    float* mol_dinv  = alloc(n_mol);
    float* proA   = alloc((size_t)n_pro * 216);   // X / conv output ping
    float* proH   = alloc((size_t)n_pro * 216);   // H = X*W
    float* proAcc = alloc((size_t)n_pro * 216);   // neighbor accumulator
    float* molA   = alloc((size_t)n_mol * 312);
    float* molH   = alloc((size_t)n_mol * 312);
    float* molAcc = alloc((size_t)n_mol * 312);
    float* mol_pool = alloc((size_t)B * 312);
    float* pro_pool = alloc((size_t)B * 216);
    float* mol_cnt  = alloc(B);
    float* pro_cnt  = alloc(B);
    float* concatb  = alloc((size_t)B * 256);     // [xt(128) | x(128)]
    float* fcb1     = alloc((size_t)B * 1024);
    float* fcb2     = alloc((size_t)B * 512);
    (void)ws_size; (void)n_in; (void)out_size;

    // ---- symmetric-norm degree terms: dinv = rsqrt(1 + in_degree) ----
    auto make_dinv = [&](const int* dstp, int ne, float* buf, int n) {
        fill1_f32<<<(n + 255) / 256, 256, 0, stream>>>(buf, n);
        deg_accum<<<(ne + 255) / 256, 256, 0, stream>>>(dstp, ne, buf);
        rsqrt_inplace<<<(n + 255) / 256, 256, 0, stream>>>(buf, n);
    };
    make_dinv(pro_oei + ne_pro, ne_pro, pro_dinv1, n_pro);  // conv1 uses orig edges
    make_dinv(pro_ei  + ne_pro, ne_pro, pro_dinv2, n_pro);  // conv2/3
    make_dinv(mol_ei  + ne_mol, ne_mol, mol_dinv,  n_mol);

    // ---- molecule branch ----
    run_gcn(mol_x, prm(0), prm(1), mol_ei, mol_ei + ne_mol, ne_mol, mol_dinv,
            molH, molAcc, molA, n_mol, 78, 78, stream);
    run_gcn(molA, prm(2), prm(3), mol_ei, mol_ei + ne_mol, ne_mol, mol_dinv,
            molH, molAcc, molA, n_mol, 78, 156, stream);
    run_gcn(molA, prm(4), prm(5), mol_ei, mol_ei + ne_mol, ne_mol, mol_dinv,
            molH, molAcc, molA, n_mol, 156, 312, stream);
    zero_f32<<<(unsigned)(((size_t)B * 312 + 255) / 256), 256, 0, stream>>>(mol_pool, (size_t)B * 312);
    zero_f32<<<1, 256, 0, stream>>>(mol_cnt, B);
    {
        dim3 ga(n_mol, (312 + 63) / 64), gd(B, (312 + 63) / 64);
        pool_accum<<<ga, 64, 0, stream>>>(molA, mol_batch, mol_pool, mol_cnt, n_mol, 312);
        pool_div<<<gd, 64, 0, stream>>>(mol_pool, mol_cnt, 312);
    }
    run_linear(mol_pool, prm(6), prm(7), fcb1, B, 312, 1024, 1024, /*bias+relu*/1, stream);
    run_linear(fcb1, prm(8), prm(9), concatb + 128, B, 1024, 128, 256, /*bias*/2, stream);

    // ---- protein branch ----
    run_gcn(pro_x, prm(10), prm(11), pro_oei, pro_oei + ne_pro, ne_pro, pro_dinv1,
            proH, proAcc, proA, n_pro, 54, 54, stream);
    run_gcn(proA, prm(12), prm(13), pro_ei, pro_ei + ne_pro, ne_pro, pro_dinv2,
            proH, proAcc, proA, n_pro, 54, 108, stream);
    run_gcn(proA, prm(14), prm(15), pro_ei, pro_ei + ne_pro, ne_pro, pro_dinv2,
            proH, proAcc, proA, n_pro, 108, 216, stream);
    zero_f32<<<(unsigned)(((size_t)B * 216 + 255) / 256), 256, 0, stream>>>(pro_pool, (size_t)B * 216);
    zero_f32<<<1, 256, 0, stream>>>(pro_cnt, B);
    {
        dim3 ga(n_pro, (216 + 63) / 64), gd(B, (216 + 63) / 64);
        pool_accum<<<ga, 64, 0, stream>>>(proA, pro_batch, pro_pool, pro_cnt, n_pro, 216);
        pool_div<<<gd, 64, 0, stream>>>(pro_pool, pro_cnt, 216);
    }
    run_linear(pro_pool, prm(16), prm(17), fcb1, B, 216, 1024, 1024, /*bias+relu*/1, stream);
    run_linear(fcb1, prm(18), prm(19), concatb + 0, B, 1024, 128, 256, /*bias*/2, stream);

    // ---- head: concat(protein, molecule) -> 1024 -> 512 -> 1 ----
    run_linear(concatb, prm(20), prm(21), fcb1, B, 256, 1024, 1024, /*bias+relu*/1, stream);
    run_linear(fcb1,    prm(22), prm(23), fcb2, B, 1024, 512, 512, /*bias+relu*/1, stream);
    run_linear(fcb2,    prm(24), prm(25), (float*)d_out, B, 512, 1, 1, /*bias*/2, stream);
}